// Thalamus_8856222564632
// MI455X (gfx1250) — compile-verified
//
#include <hip/hip_runtime.h>

// ---------------------------------------------------------------------------
// Thalamus Izhikevich network on MI455X (gfx1250), fp32 end-to-end.
// GEMVs use V_WMMA_F32_16X16X4_F32 (wave32): A = 16x4 fp32 weight tile,
// B = x broadcast across 16 columns -> every column of D is y-partial.
// Deterministic: K-sliced partials written uniquely, combined by one block.
// Shapes are template params so wave->tile decomposition is shifts, not idiv.
// ---------------------------------------------------------------------------

typedef float v2f __attribute__((ext_vector_type(2)));
typedef float v8f __attribute__((ext_vector_type(8)));

#define N_TC  4096
#define N_TRN 1024

// workspace layout (float offsets)
enum : int {
  OFF_V_TC      = 0,
  OFF_U_TC      = 4096,
  OFF_R_TC      = 8192,
  OFF_V_TRN     = 12288,
  OFF_U_TRN     = 13312,
  OFF_R_TRN     = 14336,
  OFF_ISTAT_TC  = 15360,               // raw static tect->TC current
  OFF_ISTAT_TRN = 19456,               // raw static pal->TRN current
  OFF_DYN_TC    = 20480,               // W_trn_tc @ r_trn (raw)
  OFF_DYN_TRN   = 24576,               // W_tc_trn @ r_tc  (raw)
  OFF_SCAL      = 25600,               // [0]=sum|stat_tc| [1]=sum|stat_trn| [2]=sum|dyn_trn| [3]=sum|dyn_tc|
  OFF_PSTAT_TC  = 25616,               // 8 slices * 4096
  OFF_PSTAT_TRN = 25616 + 32768,       // 8 slices * 1024
  OFF_PDYN_TRN  = 25616,               // reuses static partial space: 4 * 1024
  OFF_PDYN_TC   = 25616 + 4096         //                              2 * 4096
};
// total ws: 66576 floats (~266 KB)

// y_part[kslice*N + row] = W[row, k0:k0+KCHUNK] . x[k0:k0+KCHUNK]
// one wave handles a 16-row tile over its K slice using V_WMMA_F32_16X16X4_F32
template <int N, int K, int KSPLIT>
__global__ void gemv_wmma_f32(const float* __restrict__ W,
                              const float* __restrict__ x,
                              float* __restrict__ ypart) {
  constexpr int NTILES = N >> 4;        // power of two in all instantiations
  constexpr int KCHUNK = K / KSPLIT;    // multiple of 8
  const int lane = threadIdx.x & 31;
  const int wave = (int)((blockIdx.x * blockDim.x + threadIdx.x) >> 5);
  const int rowTile = wave & (NTILES - 1);   // wave % NTILES  (shift/mask)
  const int kslice  = wave >> __builtin_ctz(NTILES);
  const int k0      = kslice * KCHUNK;

  // A-fragment addressing per ISA 16x4 f32 layout:
  //   lanes 0-15 : row M=lane,     columns {k, k+1}
  //   lanes16-31 : row M=lane-16,  columns {k+2, k+3}
  const int row = (rowTile << 4) + (lane & 15);
  const int cb  = (lane >> 4) << 1;          // 0 or 2
  const float* Wp = W + (size_t)row * (size_t)K + (size_t)(k0 + cb);
  const float* xp = x + k0 + cb;

  v8f acc0 = {};
  v8f acc1 = {};
#pragma unroll 2
  for (int k = 0; k < KCHUNK; k += 8) {
    v2f a0 = *(const v2f*)(Wp + k);
    v2f b0 = *(const v2f*)(xp + k);
    v2f a1 = *(const v2f*)(Wp + k + 4);
    v2f b1 = *(const v2f*)(xp + k + 4);
    __builtin_prefetch((const void*)(Wp + k + 1024), 0, 0);  // global_prefetch ahead of W stream
    acc0 = __builtin_amdgcn_wmma_f32_16x16x4_f32(false, a0, false, b0,
                                                 (short)0, acc0, false, false);
    acc1 = __builtin_amdgcn_wmma_f32_16x16x4_f32(false, a1, false, b1,
                                                 (short)0, acc1, false, false);
  }
  v8f acc = acc0 + acc1;   // all columns of D identical; take column N=0

  // lane 0 holds D[0..7][0]  -> rows rowTile*16 + 0..7
  // lane 16 holds D[8..15][0] -> rows rowTile*16 + 8..15
  if ((lane & 15) == 0) {
    float* out = ypart + (size_t)kslice * (size_t)N + (rowTile << 4) + ((lane >> 4) << 3);
#pragma unroll
    for (int j = 0; j < 8; ++j) out[j] = acc[j];
  }
}

// Single-workgroup: sum K-slice partials into y, and produce sum(|y|) scalars.
// Deterministic (fixed summation order, no atomics).
__global__ void combine_reduce(const float* __restrict__ pA, float* __restrict__ yA,
                               int nA, int sA, float* __restrict__ sumA,
                               const float* __restrict__ pB, float* __restrict__ yB,
                               int nB, int sB, float* __restrict__ sumB) {
  __shared__ float red[1024];
  const int t = threadIdx.x;

  float local = 0.0f;
  for (int i = t; i < nA; i += 1024) {
    float s = 0.0f;
    for (int j = 0; j < sA; ++j) s += pA[(size_t)j * nA + i];
    yA[i] = s;
    local += fabsf(s);
  }
  red[t] = local; __syncthreads();
  for (int off = 512; off > 0; off >>= 1) {
    if (t < off) red[t] += red[t + off];
    __syncthreads();
  }
  if (t == 0) *sumA = red[0];
  __syncthreads();

  local = 0.0f;
  for (int i = t; i < nB; i += 1024) {
    float s = 0.0f;
    for (int j = 0; j < sB; ++j) s += pB[(size_t)j * nB + i];
    yB[i] = s;
    local += fabsf(s);
  }
  red[t] = local; __syncthreads();
  for (int off = 512; off > 0; off >>= 1) {
    if (t < off) red[t] += red[t + off];
    __syncthreads();
  }
  if (t == 0) *sumB = red[0];
}

__global__ void init_state(float* __restrict__ ws) {
  const int tid = blockIdx.x * blockDim.x + threadIdx.x;
  if (tid < N_TC) {
    ws[OFF_V_TC + tid] = -65.0f;
    ws[OFF_U_TC + tid] = 0.25f * -65.0f;   // LTS b*v
    ws[OFF_R_TC + tid] = 0.0f;
  }
  if (tid < N_TRN) {
    ws[OFF_V_TRN + tid] = -65.0f;
    ws[OFF_U_TRN + tid] = 0.2f * -65.0f;   // FS b*v
    ws[OFF_R_TRN + tid] = 0.0f;
  }
  if (tid < 8) ws[OFF_SCAL + tid] = 0.0f;
}

// Fused Izhikevich update for both populations (currents from pre-update rates).
__global__ void izh_update(float* __restrict__ ws, float* __restrict__ out, int write_out) {
  const float* scal = ws + OFF_SCAL;
  // _nd scale factors: m = mean|r| + 1e-8 ; scale = m > 1e-3 ? tgt/m : 0
  const float m_stc  = scal[0] * (1.0f / 4096.0f) + 1e-8f;
  const float s_stc  = (m_stc  > 1e-3f) ? 6.0f / m_stc  : 0.0f;
  const float m_strn = scal[1] * (1.0f / 1024.0f) + 1e-8f;
  const float s_strn = (m_strn > 1e-3f) ? 2.0f / m_strn : 0.0f;
  const float m_dtrn = scal[2] * (1.0f / 1024.0f) + 1e-8f;
  const float s_dtrn = (m_dtrn > 1e-3f) ? 3.0f / m_dtrn : 0.0f;
  const float m_dtc  = scal[3] * (1.0f / 4096.0f) + 1e-8f;
  const float s_dtc  = (m_dtc  > 1e-3f) ? 2.0f / m_dtc  : 0.0f;

  const int tid = blockIdx.x * blockDim.x + threadIdx.x;
  if (tid < N_TRN) {
    const int i = tid;                               // FS: a=0.1 b=0.2 c=-65 d=2, tonic 0.5
    float v = ws[OFF_V_TRN + i], u = ws[OFF_U_TRN + i], r = ws[OFF_R_TRN + i];
    float I = ws[OFF_DYN_TRN + i] * s_dtrn + ws[OFF_ISTAT_TRN + i] * s_strn;
    float vn = v + (0.04f * v * v + 5.0f * v + 140.0f - u + I + 0.5f);
    float un = u + 0.1f * (0.2f * v - u);
    float spike = (vn >= 30.0f) ? 1.0f : 0.0f;
    float vo = (spike > 0.0f) ? -65.0f : vn;
    float uo = un + 2.0f * spike;
    float ro = 0.9f * r + 0.1f * spike;
    ws[OFF_V_TRN + i] = vo; ws[OFF_U_TRN + i] = uo; ws[OFF_R_TRN + i] = ro;
    if (write_out) out[4096 + i] = ro;               // trn_rate
  } else if (tid < N_TRN + N_TC) {
    const int i = tid - N_TRN;                       // LTS: a=0.02 b=0.25 c=-65 d=2, tonic 0, na_bias 1.5
    float v = ws[OFF_V_TC + i], u = ws[OFF_U_TC + i], r = ws[OFF_R_TC + i];
    float I = ws[OFF_ISTAT_TC + i] * s_stc - ws[OFF_DYN_TC + i] * s_dtc + 1.5f;
    float vn = v + (0.04f * v * v + 5.0f * v + 140.0f - u + I + 0.0f);
    float un = u + 0.02f * (0.25f * v - u);
    float spike = (vn >= 30.0f) ? 1.0f : 0.0f;
    float vo = (spike > 0.0f) ? -65.0f : vn;
    float uo = un + 2.0f * spike;
    float ro = 0.9f * r + 0.1f * spike;
    ws[OFF_V_TC + i] = vo; ws[OFF_U_TC + i] = uo; ws[OFF_R_TC + i] = ro;
    if (write_out) { out[i] = ro; out[5120 + i] = ro; }  // tc_rate, prediction_error
  }
}

extern "C" void kernel_launch(void* const* d_in, const int* in_sizes, int n_in,
                              void* d_out, int out_size, void* d_ws, size_t ws_size,
                              hipStream_t stream) {
  (void)in_sizes; (void)n_in; (void)out_size; (void)ws_size;
  const float* tect      = (const float*)d_in[0];  // [8192]
  const float* pal       = (const float*)d_in[1];  // [8192]
  const float* W_tect_tc = (const float*)d_in[2];  // [4096, 8192]
  const float* W_pal_trn = (const float*)d_in[3];  // [1024, 8192]
  const float* W_tc_trn  = (const float*)d_in[4];  // [1024, 4096]
  const float* W_trn_tc  = (const float*)d_in[5];  // [4096, 1024]
  float* out = (float*)d_out;                      // [4096 + 1024 + 4096]
  float* ws  = (float*)d_ws;

  init_state<<<16, 256, 0, stream>>>(ws);

  // Loop-invariant GEMVs (inputs never change across substeps): compute once.
  // W_tect_tc: 256 row tiles * 8 K-slices = 2048 waves -> 256 blocks of 256
  gemv_wmma_f32<4096, 8192, 8><<<256, 256, 0, stream>>>(W_tect_tc, tect, ws + OFF_PSTAT_TC);
  // W_pal_trn: 64 row tiles * 8 K-slices = 512 waves -> 64 blocks
  gemv_wmma_f32<1024, 8192, 8><<<64, 256, 0, stream>>>(W_pal_trn, pal, ws + OFF_PSTAT_TRN);
  combine_reduce<<<1, 1024, 0, stream>>>(ws + OFF_PSTAT_TC,  ws + OFF_ISTAT_TC,  4096, 8, ws + OFF_SCAL + 0,
                                         ws + OFF_PSTAT_TRN, ws + OFF_ISTAT_TRN, 1024, 8, ws + OFF_SCAL + 1);

  for (int s = 0; s < 8; ++s) {
    // dynamic GEMVs from previous-substep rates (16 MB each -> L2 resident)
    // W_tc_trn @ r_tc: 64 tiles * 4 slices = 256 waves -> 32 blocks
    gemv_wmma_f32<1024, 4096, 4><<<32, 256, 0, stream>>>(W_tc_trn, ws + OFF_R_TC, ws + OFF_PDYN_TRN);
    // W_trn_tc @ r_trn: 256 tiles * 2 slices = 512 waves -> 64 blocks
    gemv_wmma_f32<4096, 1024, 2><<<64, 256, 0, stream>>>(W_trn_tc, ws + OFF_R_TRN, ws + OFF_PDYN_TC);
    combine_reduce<<<1, 1024, 0, stream>>>(ws + OFF_PDYN_TRN, ws + OFF_DYN_TRN, 1024, 4, ws + OFF_SCAL + 2,
                                           ws + OFF_PDYN_TC,  ws + OFF_DYN_TC,  4096, 2, ws + OFF_SCAL + 3);
    izh_update<<<20, 256, 0, stream>>>(ws, out, (s == 7) ? 1 : 0);
  }
}